// PCDRender_42030549959314
// MI455X (gfx1250) — compile-verified
//
#include <hip/hip_runtime.h>
#include <stdint.h>

#ifndef __has_builtin
#define __has_builtin(x) 0
#endif

#define AS1 __attribute__((address_space(1)))
#define AS3 __attribute__((address_space(3)))

typedef __attribute__((ext_vector_type(2))) float v2f;
typedef __attribute__((ext_vector_type(4))) float v4f;
typedef __attribute__((ext_vector_type(8))) float v8f;
typedef __attribute__((ext_vector_type(4))) int   v4i;

static constexpr int   kH = 500, kW = 512, kK = 10, kN = 8192;
static constexpr int   kTiles  = kN / 16;              // 512 point tiles
static constexpr float kR2     = 0.005f * 0.005f;      // 2.5e-5
static constexpr float kInvR2  = 1.0f / kR2;
static constexpr float kFx = 575.f, kFy = 575.f, kPx = 256.f, kPy = 250.f;
static constexpr float kInvSmin = 1.0f / 500.0f;
static constexpr int   kApkFloats = kTiles * 64;       // 32768 floats (128 KB) WMMA A-operand table
static constexpr int   kWsFloats  = kApkFloats + kN;   // + z table = 40960 floats (160 KB)

#if __has_builtin(__builtin_amdgcn_global_load_async_to_lds_b128)
#define HAVE_ASYNC_LDS 1
#else
#define HAVE_ASYNC_LDS 0
#endif

#if __has_builtin(__builtin_amdgcn_s_wait_asynccnt)
#define ASYNC_WAIT0() __builtin_amdgcn_s_wait_asynccnt(0)
#else
#define ASYNC_WAIT0() asm volatile("s_wait_asynccnt 0x0" ::: "memory")
#endif

// ---------------------------------------------------------------------------
// Pre-pass: project points, build the WMMA A-operand table in d_ws.
// A is 16(points) x 4(K) f32. Register layout (ISA 7.12.2):
//   lane m    (m<16): v0 = K0, v1 = K1
//   lane m+16       : v0 = K2, v1 = K3
// Stored per tile t as [32 lanes][2 floats] so the render kernel does one
// ds_load_b64 per lane per tile:
//   lane m   : (-2*x_ndc, -2*y_ndc)
//   lane m+16: (x_ndc^2 + y_ndc^2, 1.0)
// ---------------------------------------------------------------------------
__global__ __launch_bounds__(256)
void pcd_prep_kernel(const float* __restrict__ pts, float* __restrict__ ws) {
  const int j = blockIdx.x * blockDim.x + threadIdx.x;
  if (j >= kN) return;
  const float x = pts[j * 3 + 0];
  const float y = pts[j * 3 + 1];
  const float z = pts[j * 3 + 2];
  const float invz = 1.0f / z;
  const float u = -kFx * x * invz + kPx;
  const float v = -kFy * y * invz + kPy;
  const float xn = -(2.0f * u - (float)kW) * kInvSmin;
  const float yn = -(2.0f * v - (float)kH) * kInvSmin;
  const int t = j >> 4, l = j & 15;
  float* apk = ws + t * 64;
  apk[l * 2 + 0]        = -2.0f * xn;
  apk[l * 2 + 1]        = -2.0f * yn;
  apk[(l + 16) * 2 + 0] = xn * xn + yn * yn;
  apk[(l + 16) * 2 + 1] = 1.0f;
  ws[kApkFloats + j] = z;
}

// Branchless sorted insert (ascending z), keeps the 10 smallest. Fully
// unrolled so zl/dl/il stay in VGPRs (no dynamic indexing -> no scratch).
__device__ __forceinline__ void insert10(float z, float d2, int idx,
                                         float (&zl)[kK], float (&dl)[kK],
                                         int (&il)[kK]) {
  float pz = z, pd = d2;
  int   pi = idx;
#pragma unroll
  for (int i = 0; i < kK; ++i) {
    const bool  sw = pz < zl[i];
    const float tz = zl[i], td = dl[i];
    const int   ti = il[i];
    zl[i] = sw ? pz : tz;  dl[i] = sw ? pd : td;  il[i] = sw ? pi : ti;
    pz = sw ? tz : pz;     pd = sw ? td : pd;     pi = sw ? ti : pi;
  }
}

// ---------------------------------------------------------------------------
// Main kernel: one wave = 16 pixels (2 lanes per pixel). Per tile:
// ds_load_b64 prefetch of A(t+1), one v_wmma_f32_16x16x4_f32 (16x16 d^2
// tile), v_min3 screening tree, rare exec-masked top-k insertion.
// ---------------------------------------------------------------------------
__global__ __launch_bounds__(256)
void pcd_render_kernel(const float* __restrict__ ws,
                       const float* __restrict__ feat,
                       float* __restrict__ img,
                       float* __restrict__ zbuf) {
  extern __shared__ float smem[];  // kWsFloats (160 KB): Apk table + z table
  const int tid = threadIdx.x;

  // ---- stage point tables into LDS (async copy path on gfx1250) ----
#if HAVE_ASYNC_LDS
  {
    AS1 char* g = (AS1 char*)ws;
    AS3 char* l = (AS3 char*)smem;
    for (int ofs = tid * 16; ofs < kWsFloats * 4; ofs += 256 * 16)
      __builtin_amdgcn_global_load_async_to_lds_b128(
          (AS1 v4i*)(g + ofs), (AS3 v4i*)(l + ofs), 0, 0);
    ASYNC_WAIT0();
  }
#else
  {
    const v4f* g4 = (const v4f*)ws;
    v4f*       l4 = (v4f*)smem;
    for (int i = tid; i < kWsFloats / 4; i += 256) l4[i] = g4[i];
  }
#endif
  __syncthreads();

  const int lane = tid & 31;
  const int wv   = tid >> 5;
  const int nIdx = lane & 15;                       // pixel slot within wave
  const int row  = blockIdx.x >> 2;                 // 500 rows
  const int col  = (blockIdx.x & 3) * 128 + wv * 16 + nIdx;

  // Pixel NDC coords: xs = (W - 2(c+0.5))/S_MIN, ys = (H - 2(r+0.5))/S_MIN
  const float px = ((float)kW - 2.0f * ((float)col + 0.5f)) * kInvSmin;
  const float py = ((float)kH - 2.0f * ((float)row + 0.5f)) * kInvSmin;
  // B operand (4x16 f32): lanes 0-15 hold K0/K1 rows, lanes 16-31 K2/K3.
  v2f B;
  if (lane < 16) { B[0] = px;   B[1] = py; }
  else           { B[0] = 1.0f; B[1] = px * px + py * py; }

  float zl[kK], dl[kK];
  int   il[kK];
#pragma unroll
  for (int i = 0; i < kK; ++i) { zl[i] = 3.0e38f; dl[i] = 0.0f; il[i] = 0; }

  const v2f*   apk = (const v2f*)smem;        // [kTiles*32] lane-ready A ops
  const float* zt  = smem + kApkFloats;       // [kN] point depths
  const v8f    czero = {};
  const int    mOfs = (lane < 16) ? 0 : 8;    // C rows owned by this lane

  // A operand of tile 0, then keep one tile of prefetch distance so the
  // ds_load and its consuming WMMA are an iteration apart (dscnt <= 1).
  v2f Acur = apk[lane];

  for (int t = 0; t < kTiles; ++t) {
    const v2f Apre = apk[((t + 1) & (kTiles - 1)) * 32 + lane];
    // d2[e]: pixel = this lane, point = t*16 + mOfs + e
    const v8f d2 = __builtin_amdgcn_wmma_f32_16x16x4_f32(
        false, Acur, false, B, (short)0, czero, false, false);

    const float m01 = fminf(fminf(d2[0], d2[1]), fminf(d2[2], d2[3]));
    const float m23 = fminf(fminf(d2[4], d2[5]), fminf(d2[6], d2[7]));
    if (fminf(m01, m23) < kR2) {              // rare hit path
      const int mb = t * 16 + mOfs;
#pragma unroll
      for (int e = 0; e < 8; ++e) {
        const float de = d2[e];
        if (de < kR2) {
          const int   m = mb + e;
          const float z = zt[m];
          if (z > 0.0f) insert10(z, de, m, zl, dl, il);
        }
      }
    }
    Acur = Apre;
  }

  // Merge the two half-lists of each pixel (lane j <-> lane j+16).
#pragma unroll
  for (int k = 0; k < kK; ++k) {
    const float oz = __shfl_xor(zl[k], 16, 32);
    const float od = __shfl_xor(dl[k], 16, 32);
    const int   oi = __shfl_xor(il[k], 16, 32);
    insert10(oz, od, oi, zl, dl, il);         // INF entries pass through
  }

  // Composite + store (one lane per pixel).
  if (lane < 16) {
    const size_t pix = (size_t)row * kW + col;
    float cum = 1.0f, a0 = 0.0f, a1 = 0.0f, a2 = 0.0f;
    float* zb = zbuf + pix * kK;
#pragma unroll
    for (int k = 0; k < kK; ++k) {
      const bool  ok = zl[k] < 1.0e37f;
      const float w  = ok ? (1.0f - dl[k] * kInvR2) : 0.0f;
      const float cw = w * cum;
      const int   fi = il[k] * 3;             // il defaults to 0 -> safe load
      a0 += cw * feat[fi + 0];
      a1 += cw * feat[fi + 1];
      a2 += cw * feat[fi + 2];
      cum *= (1.0f - w);
      zb[k] = ok ? zl[k] : -1.0f;
    }
    float* ip = img + pix * 3;
    ip[0] = a0; ip[1] = a1; ip[2] = a2;
  }
}

extern "C" void kernel_launch(void* const* d_in, const int* in_sizes, int n_in,
                              void* d_out, int out_size, void* d_ws,
                              size_t ws_size, hipStream_t stream) {
  (void)in_sizes; (void)n_in; (void)out_size; (void)ws_size;
  const float* pts  = (const float*)d_in[0];
  const float* feat = (const float*)d_in[1];
  float*       out  = (float*)d_out;
  float*       ws   = (float*)d_ws;

  pcd_prep_kernel<<<kN / 256, 256, 0, stream>>>(pts, ws);

  const size_t smemBytes = (size_t)kWsFloats * sizeof(float);  // 160 KB
  (void)hipFuncSetAttribute(reinterpret_cast<const void*>(pcd_render_kernel),
                            hipFuncAttributeMaxDynamicSharedMemorySize,
                            (int)smemBytes);
  // grid = 500 rows * 4 column segments; block = 256 threads = 8 waves.
  pcd_render_kernel<<<kH * (kW / 128), 256, smemBytes, stream>>>(
      ws, feat, out, out + (size_t)kH * kW * 3);
}